// StructuralCTCLoss_24361054503518
// MI455X (gfx1250) — compile-verified
//
#include <hip/hip_runtime.h>
#include <stdint.h>

// Problem constants (match the reference).
#define T_DIM 1024
#define N_DIM 64
#define C_DIM 512
#define S_DIM 64
#define U_DIM 129          // 2*S + 1
#define U_PAD 132          // padded row stride (16B aligned)
#define NEGV  (-1e30f)

typedef __attribute__((ext_vector_type(2))) float v2f;
typedef __attribute__((ext_vector_type(8))) float v8f;

// ---------------------------------------------------------------------------
// flat 3-way log-sum-exp (stable: sum in [1,3], exp underflow of -1e30 -> 0)
// ---------------------------------------------------------------------------
__device__ __forceinline__ float lae3(float x, float y, float z) {
    float m = fmaxf(fmaxf(x, y), z);
    float s = __expf(x - m) + __expf(y - m) + __expf(z - m);
    return m + __logf(s);
}

// ---------------------------------------------------------------------------
// Kernel 0: build extended-label table ext[n][u] and allow_skip[n][u].
// ---------------------------------------------------------------------------
__global__ void k_ext(const int* __restrict__ targets,
                      int* __restrict__ ext, int* __restrict__ allow) {
    int i = blockIdx.x * 256 + threadIdx.x;
    if (i >= N_DIM * U_PAD) return;
    int n = i / U_PAD, u = i % U_PAD;
    int e = 0, a = 0;
    if (u < U_DIM) {
        e = (u & 1) ? targets[n * S_DIM + (u >> 1)] : 0;
        int e2 = -1;                                   // ext2 = -1 for u < 2
        if (u >= 2) e2 = ((u - 2) & 1) ? targets[n * S_DIM + ((u - 2) >> 1)] : 0;
        a = (e != 0 && e != e2) ? 1 : 0;
    }
    ext[n * U_PAD + u]   = e;
    allow[n * U_PAD + u] = a;
}

// ---------------------------------------------------------------------------
// Kernel A: one wave32 per (t,n) row of log_probs.
//  - argmax over C=512 (16 values/lane via 4x b128 loads, shfl_xor reduce,
//    first-index tie-break to match jnp.argmax)
//  - GATHER: park the row in LDS, emit lp[t,n,ext[n,u]] for u<129 to lpbuf
//    so the CTC scan never re-reads the 128 MiB tensor.
// ---------------------------------------------------------------------------
template <bool GATHER>
__global__ void k_argmax(const float* __restrict__ logp,
                         const int* __restrict__ ext,
                         int* __restrict__ pred,
                         float* __restrict__ lpbuf) {
    __shared__ float lds[8 * C_DIM];                   // 16 KB: 8 waves/block
    int wave = threadIdx.x >> 5, lane = threadIdx.x & 31;
    int row = blockIdx.x * 8 + wave;                   // row in [0, T*N)
    int t = row / N_DIM, n = row % N_DIM;

    const float4* rp = (const float4*)(logp + ((size_t)t * N_DIM + n) * C_DIM);
    float* lrow = lds + wave * C_DIM;

    float best = NEGV;
    int bidx = 0;
#pragma unroll
    for (int k = 0; k < 4; ++k) {                      // coalesced 512B/load
        float4 v = rp[lane + 32 * k];
        if (GATHER) ((float4*)lrow)[lane + 32 * k] = v;
        int c0 = (lane + 32 * k) * 4;
        if (v.x > best) { best = v.x; bidx = c0; }
        if (v.y > best) { best = v.y; bidx = c0 + 1; }
        if (v.z > best) { best = v.z; bidx = c0 + 2; }
        if (v.w > best) { best = v.w; bidx = c0 + 3; }
    }
#pragma unroll
    for (int off = 16; off > 0; off >>= 1) {           // wave32 argmax reduce
        float ov = __shfl_xor(best, off);
        int   oi = __shfl_xor(bidx, off);
        if (ov > best || (ov == best && oi < bidx)) { best = ov; bidx = oi; }
    }
    if (lane == 0) pred[t * N_DIM + n] = bidx;         // [T][N] for coalesced scan

    if (GATHER) {
        __builtin_amdgcn_wave_barrier();               // order LDS store->load
        const int* extn = ext + n * U_PAD;
        float* orow = lpbuf + ((size_t)n * T_DIM + t) * U_PAD;
#pragma unroll
        for (int j = 0; j < 5; ++j) {
            int u = lane + 32 * j;
            if (u < U_DIM) orow[u] = lrow[extn[u]];
        }
    }
}

// ---------------------------------------------------------------------------
// Kernel B: CTC forward recursion. One wave32 per batch row n; lane l owns
// states u = 5l..5l+4 (26 active lanes cover U=129). Per step: 2 shfl_up
// (a[u-1], a[u-2] across the lane boundary), 5 independent logaddexp3 (ILP),
// no barriers. T is serial by nature -> latency-optimized inner loop.
// ---------------------------------------------------------------------------
template <bool GATHER>
__global__ void k_ctc(const float* __restrict__ logp,
                      const float* __restrict__ lpbuf,
                      const int* __restrict__ ext,
                      const int* __restrict__ allow,
                      const int* __restrict__ IL,
                      const int* __restrict__ TL,
                      float* __restrict__ content) {
    __shared__ float sv[2];
    int n = blockIdx.x, l = threadIdx.x;
    int u0 = 5 * l;

    int extv[5]; int alw[5];
#pragma unroll
    for (int j = 0; j < 5; ++j) {
        int u = u0 + j;
        bool in = (u < U_DIM);
        extv[j] = in ? ext[n * U_PAD + u]   : 0;
        alw[j]  = in ? allow[n * U_PAD + u] : 0;
    }
    int Ln = IL[n];

    float A[5];
    {   // t = 0 init: a0[0]=lp[0,0], a0[1]=lp[0,1], else NEG
#pragma unroll
        for (int j = 0; j < 5; ++j) {
            int u = u0 + j;
            float v = NEGV;
            if (u < 2) {
                v = GATHER ? lpbuf[((size_t)n * T_DIM + 0) * U_PAD + u]
                           : logp[((size_t)0 * N_DIM + n) * C_DIM + extv[j]];
            }
            A[j] = v;
        }
    }

    for (int t = 1; t < Ln; ++t) {
        float pA4 = __shfl_up(A[4], 1);                // a[5l-1] from lane l-1
        float pA3 = __shfl_up(A[3], 1);                // a[5l-2] from lane l-1
        if (l == 0) { pA4 = NEGV; pA3 = NEGV; }

        float lpt[5];
#pragma unroll
        for (int j = 0; j < 5; ++j) {
            int u = u0 + j;
            float v = NEGV;
            if (u < U_DIM) {
                v = GATHER ? lpbuf[((size_t)n * T_DIM + t) * U_PAD + u]
                           : logp[((size_t)t * N_DIM + n) * C_DIM + extv[j]];
            }
            lpt[j] = v;
        }

        float Nw[5];
#pragma unroll
        for (int j = 0; j < 5; ++j) {
            float am1 = (j >= 1) ? A[j - 1] : pA4;
            float am2 = (j >= 2) ? A[j - 2] : ((j == 1) ? pA4 : pA3);
            float a2v = alw[j] ? am2 : NEGV;
            Nw[j] = lae3(A[j], am1, a2v) + lpt[j];
        }
#pragma unroll
        for (int j = 0; j < 5; ++j) A[j] = Nw[j];
    }

    int ul = 2 * TL[n];                                // u_last in [64,128]
#pragma unroll
    for (int j = 0; j < 5; ++j) {
        int u = u0 + j;
        if (u == ul)     sv[1] = A[j];
        if (u == ul - 1) sv[0] = A[j];
    }
    __syncthreads();
    if (l == 0) {
        float x = sv[1], y = sv[0];
        float m = fmaxf(x, y);
        float loss = -(m + log1pf(__expf(fminf(x, y) - m)));   // -logaddexp
        content[n] = (loss > 1e20f) ? 0.0f : loss;
    }
}

// ---------------------------------------------------------------------------
// Kernel P: struct penalty. Thread n scans pred[t][n] (coalesced across the
// 64 threads), compacting struct tokens (1..8, dedup against prev, non-blank)
// and comparing against the compacted target struct sequence.
// ---------------------------------------------------------------------------
__global__ void k_penalty(const int* __restrict__ pred,
                          const int* __restrict__ targets,
                          const int* __restrict__ IL,
                          const int* __restrict__ TL,
                          float* __restrict__ penalty) {
    __shared__ int tseq[N_DIM * S_DIM];                // 16 KB, private rows
    int n = threadIdx.x;
    if (n >= N_DIM) return;

    int tl = TL[n], tcnt = 0;
    for (int s = 0; s < S_DIM; ++s) {
        int tok = targets[n * S_DIM + s];
        if (s < tl && tok >= 1 && tok <= 8) tseq[n * S_DIM + tcnt++] = tok;
    }
    int Ln = IL[n];
    int prev = -1, pcnt = 0, mism = 0;
    for (int t = 0; t < Ln; ++t) {                     // beyond Ln: p==BLANK, no keeps
        int p = pred[t * N_DIM + n];
        if (p != prev && p != 0 && p <= 8) {           // p>=1 implied by !=0 & argmax>=0
            if (p >= 1) {
                if (pcnt >= tcnt || tseq[n * S_DIM + pcnt] != p) mism = 1;
                ++pcnt;
            }
        }
        prev = p;
    }
    if (pcnt != tcnt) mism = 1;
    penalty[n] = mism ? 1.0f : 0.0f;
}

// ---------------------------------------------------------------------------
// Kernel C: total = content + penalty; mean over N=64 via one
// V_WMMA_F32_16X16X4_F32: A = losses as 16x4, B = 4x16 ones -> D rows hold
// row-sums; in-lane add of the 8 C/D VGPRs + shfl_xor(16) finishes the sum.
// ---------------------------------------------------------------------------
__global__ void k_mean(const float* __restrict__ content,
                       const float* __restrict__ penalty,
                       float* __restrict__ out) {
    int l = threadIdx.x;                               // one wave32
    // A-matrix 16x4 f32 layout: lanes 0-15 hold (M=l, K=0),(K=1);
    //                           lanes 16-31 hold (M=l-16, K=2),(K=3).
    int i0 = 4 * (l & 15) + 2 * (l >> 4);              // element (m, k) = tot[4m+k]
    v2f a;
    a.x = content[i0]     + penalty[i0];
    a.y = content[i0 + 1] + penalty[i0 + 1];
    v2f b; b.x = 1.0f; b.y = 1.0f;                     // all-ones B (layout-invariant)
    v8f c = {};
    c = __builtin_amdgcn_wmma_f32_16x16x4_f32(
            /*neg_a=*/false, a, /*neg_b=*/false, b,
            /*c_mod=*/(short)0, c, /*reuse_a=*/false, /*reuse_b=*/false);
    float s = c[0] + c[1] + c[2] + c[3] + c[4] + c[5] + c[6] + c[7];
    s += __shfl_xor(s, 16);                            // rows 0-7 + rows 8-15
    if (l == 0) out[0] = s * (1.0f / (float)N_DIM);
}

// ---------------------------------------------------------------------------
extern "C" void kernel_launch(void* const* d_in, const int* in_sizes, int n_in,
                              void* d_out, int out_size, void* d_ws, size_t ws_size,
                              hipStream_t stream) {
    const float* logp    = (const float*)d_in[0];
    const int*   targets = (const int*)d_in[1];
    const int*   IL      = (const int*)d_in[2];
    const int*   TL      = (const int*)d_in[3];
    float* out = (float*)d_out;

    char* ws = (char*)d_ws;
    size_t off = 0;
    auto alloc = [&](size_t bytes) -> void* {
        off = (off + 255) & ~(size_t)255;
        void* p = ws + off;
        off += bytes;
        return p;
    };
    int*   ext     = (int*)  alloc((size_t)N_DIM * U_PAD * 4);
    int*   allow   = (int*)  alloc((size_t)N_DIM * U_PAD * 4);
    int*   pred    = (int*)  alloc((size_t)T_DIM * N_DIM * 4);
    float* content = (float*)alloc(N_DIM * 4);
    float* penalty = (float*)alloc(N_DIM * 4);

    size_t lpbytes = (size_t)N_DIM * T_DIM * U_PAD * 4;   // ~34.6 MB
    size_t aligned = (off + 255) & ~(size_t)255;
    bool gather = (aligned + lpbytes) <= ws_size;          // fixed per run: deterministic
    float* lpbuf = gather ? (float*)alloc(lpbytes) : nullptr;

    k_ext<<<(N_DIM * U_PAD + 255) / 256, 256, 0, stream>>>(targets, ext, allow);

    int rows = T_DIM * N_DIM;                              // 65536 rows, 8 waves/block
    if (gather) {
        k_argmax<true ><<<rows / 8, 256, 0, stream>>>(logp, ext, pred, lpbuf);
        k_ctc   <true ><<<N_DIM, 32, 0, stream>>>(logp, lpbuf, ext, allow, IL, TL, content);
    } else {
        k_argmax<false><<<rows / 8, 256, 0, stream>>>(logp, ext, pred, nullptr);
        k_ctc   <false><<<N_DIM, 32, 0, stream>>>(logp, nullptr, ext, allow, IL, TL, content);
    }
    k_penalty<<<1, 64, 0, stream>>>(pred, targets, IL, TL, penalty);
    k_mean<<<1, 32, 0, stream>>>(content, penalty, out);
}